// BCEDecorrelatedLoss_17231408792190
// MI455X (gfx1250) — compile-verified
//
#include <hip/hip_runtime.h>
#include <hip/hip_bf16.h>
#include <math.h>

typedef __attribute__((ext_vector_type(2))) float v2f;
typedef __attribute__((ext_vector_type(8))) float v8f;

#define CHUNK 2048   // j-chunk staged in LDS (multiple of 256 and 4)

// ---------------------------------------------------------------------------
// Pass 1: rowsum_a[i] = sum_j |x_i - x_j| * w_j   (and same for e)
// One wave owns 16 rows; row sums accumulate in a WMMA f32 16x16x4 accumulator
// with B = all-ones (D[m][n] = sum_k A[m][k], replicated over n).
// A-tile layout (ISA 7.12.2, 32-bit A 16x4): lane l -> row m=l&15,
// v0 holds K = (l<16?0:2), v1 holds K+1.
// ---------------------------------------------------------------------------
__global__ __launch_bounds__(256) void disco_pass1(
    const float* __restrict__ x, const float* __restrict__ e,
    const float* __restrict__ w,
    float* __restrict__ rowsum_a, float* __restrict__ rowsum_b, int n)
{
    __shared__ float sx[CHUNK], se[CHUNK], sw[CHUNK];

    const int tid  = threadIdx.x;
    const int lane = tid & 31;
    const int wave = tid >> 5;
    const int m    = lane & 15;
    const int kb   = (lane >> 4) << 1;          // 0 for lanes 0-15, 2 for 16-31
    const int i0   = blockIdx.x * 128 + wave * 16;

    const float xi = x[i0 + m];
    const float ei = e[i0 + m];

    v8f acc_a = {0.f,0.f,0.f,0.f,0.f,0.f,0.f,0.f};
    v8f acc_b = {0.f,0.f,0.f,0.f,0.f,0.f,0.f,0.f};
    const v2f ones = {1.0f, 1.0f};

    for (int jc = 0; jc < n; jc += CHUNK) {
        __syncthreads();
        for (int k = tid; k < CHUNK; k += 256) {
            sx[k] = x[jc + k];
            se[k] = e[jc + k];
            sw[k] = w[jc + k];
        }
        __syncthreads();

        for (int j0 = 0; j0 < CHUNK; j0 += 4) {
            const float xj0 = sx[j0 + kb], xj1 = sx[j0 + kb + 1];
            const float ej0 = se[j0 + kb], ej1 = se[j0 + kb + 1];
            const float wj0 = sw[j0 + kb], wj1 = sw[j0 + kb + 1];

            v2f Aa, Ab;
            Aa.x = fabsf(xi - xj0) * wj0;
            Aa.y = fabsf(xi - xj1) * wj1;
            Ab.x = fabsf(ei - ej0) * wj0;
            Ab.y = fabsf(ei - ej1) * wj1;

            acc_a = __builtin_amdgcn_wmma_f32_16x16x4_f32(
                false, Aa, false, ones, (short)0, acc_a, false, false);
            acc_b = __builtin_amdgcn_wmma_f32_16x16x4_f32(
                false, Ab, false, ones, (short)0, acc_b, false, false);
        }
    }

    // C/D layout: VGPR r holds M=r (lanes 0-15) and M=r+8 (lanes 16-31);
    // all 16 columns are identical (B = ones), so read column at lane 0 / 16.
    if (lane == 0) {
        #pragma unroll
        for (int r = 0; r < 8; ++r) {
            rowsum_a[i0 + r] = acc_a[r];
            rowsum_b[i0 + r] = acc_b[r];
        }
    }
    if (lane == 16) {
        #pragma unroll
        for (int r = 0; r < 8; ++r) {
            rowsum_a[i0 + 8 + r] = acc_a[r];
            rowsum_b[i0 + 8 + r] = acc_b[r];
        }
    }
}

// ---------------------------------------------------------------------------
// Stats: S = sum(w); meanA = sum(rowsum_a*w)/S^2; meanB likewise;
// bce_mean = mean((softplus(x) - x*y)*w). Single block, deterministic tree.
// ---------------------------------------------------------------------------
__global__ __launch_bounds__(256) void disco_stats(
    const float* __restrict__ x, const float* __restrict__ y,
    const float* __restrict__ w,
    const float* __restrict__ rowsum_a, const float* __restrict__ rowsum_b,
    float* __restrict__ sc, int n)
{
    __shared__ float r0[256], r1[256], r2[256], r3[256];
    const int tid = threadIdx.x;
    float sumw = 0.f, sumbce = 0.f, suma = 0.f, sumb = 0.f;
    for (int idx = tid; idx < n; idx += 256) {
        const float wi = w[idx];
        const float xv = x[idx];
        const float yv = y[idx];
        // logaddexp(0,x) = max(x,0) + log1p(exp(-|x|))
        const float sp = fmaxf(xv, 0.f) + log1pf(expf(-fabsf(xv)));
        sumw   += wi;
        sumbce += (sp - xv * yv) * wi;
        suma   += rowsum_a[idx] * wi;
        sumb   += rowsum_b[idx] * wi;
    }
    r0[tid] = sumw; r1[tid] = sumbce; r2[tid] = suma; r3[tid] = sumb;
    __syncthreads();
    for (int s = 128; s > 0; s >>= 1) {
        if (tid < s) {
            r0[tid] += r0[tid + s];
            r1[tid] += r1[tid + s];
            r2[tid] += r2[tid + s];
            r3[tid] += r3[tid + s];
        }
        __syncthreads();
    }
    if (tid == 0) {
        const float S = r0[0];
        sc[0] = S;                        // sum of weights
        sc[1] = r2[0] / (S * S);          // meanA
        sc[2] = r3[0] / (S * S);          // meanB
        sc[3] = r1[0] / (float)n;         // bce_mean
    }
}

// ---------------------------------------------------------------------------
// Pass 2: Tab = sum_ij A*B*w_i*w_j, Taa, Tbb with
// A_ij = |x_i-x_j| - ra_j - ra_i + meanA, ra = rowsum_a/S.
// One thread per i; j staged in LDS chunks. Per-block partials to ws.
// ---------------------------------------------------------------------------
__global__ __launch_bounds__(256) void disco_pass2(
    const float* __restrict__ x, const float* __restrict__ e,
    const float* __restrict__ w,
    const float* __restrict__ rowsum_a, const float* __restrict__ rowsum_b,
    const float* __restrict__ sc, float* __restrict__ partials, int n)
{
    __shared__ float sx[CHUNK], se[CHUNK], sw[CHUNK], sra[CHUNK], srb[CHUNK];
    __shared__ float red[256];

    const int tid = threadIdx.x;
    const int i   = blockIdx.x * 256 + tid;

    const float S    = sc[0];
    const float mA   = sc[1];
    const float mB   = sc[2];
    const float invS = 1.0f / S;

    const float xi  = x[i];
    const float ei  = e[i];
    const float wi  = w[i];
    const float rai = rowsum_a[i] * invS;
    const float rbi = rowsum_b[i] * invS;

    float sab = 0.f, saa = 0.f, sbb = 0.f;

    for (int jc = 0; jc < n; jc += CHUNK) {
        __syncthreads();
        for (int k = tid; k < CHUNK; k += 256) {
            sx[k]  = x[jc + k];
            se[k]  = e[jc + k];
            sw[k]  = w[jc + k];
            sra[k] = rowsum_a[jc + k] * invS;
            srb[k] = rowsum_b[jc + k] * invS;
        }
        __syncthreads();
        for (int j = 0; j < CHUNK; ++j) {
            const float A  = fabsf(xi - sx[j]) - sra[j] - rai + mA;
            const float B  = fabsf(ei - se[j]) - srb[j] - rbi + mB;
            const float wj = sw[j];
            sab += A * B * wj;
            saa += A * A * wj;
            sbb += B * B * wj;
        }
    }
    sab *= wi; saa *= wi; sbb *= wi;

    // three deterministic tree reductions
    red[tid] = sab; __syncthreads();
    for (int s = 128; s > 0; s >>= 1) { if (tid < s) red[tid] += red[tid + s]; __syncthreads(); }
    if (tid == 0) partials[blockIdx.x * 3 + 0] = red[0];
    __syncthreads();

    red[tid] = saa; __syncthreads();
    for (int s = 128; s > 0; s >>= 1) { if (tid < s) red[tid] += red[tid + s]; __syncthreads(); }
    if (tid == 0) partials[blockIdx.x * 3 + 1] = red[0];
    __syncthreads();

    red[tid] = sbb; __syncthreads();
    for (int s = 128; s > 0; s >>= 1) { if (tid < s) red[tid] += red[tid + s]; __syncthreads(); }
    if (tid == 0) partials[blockIdx.x * 3 + 2] = red[0];
}

// ---------------------------------------------------------------------------
// Finalize: reduce block partials serially (deterministic), emit 3 outputs.
// ---------------------------------------------------------------------------
__global__ void disco_finalize(const float* __restrict__ sc,
                               const float* __restrict__ partials,
                               int nblocks, float* __restrict__ out)
{
    if (threadIdx.x == 0 && blockIdx.x == 0) {
        float Tab = 0.f, Taa = 0.f, Tbb = 0.f;
        for (int b = 0; b < nblocks; ++b) {
            Tab += partials[b * 3 + 0];
            Taa += partials[b * 3 + 1];
            Tbb += partials[b * 3 + 2];
        }
        const float disco = Tab / sqrtf(Taa * Tbb);  // scale factors cancel
        const float bce   = sc[3];
        out[0] = bce;
        out[1] = disco;
        out[2] = bce + 0.1f * disco;
    }
}

extern "C" void kernel_launch(void* const* d_in, const int* in_sizes, int n_in,
                              void* d_out, int out_size, void* d_ws, size_t ws_size,
                              hipStream_t stream) {
    const float* x = (const float*)d_in[0];   // outputs (logits)
    const float* y = (const float*)d_in[1];   // labels
    const float* e = (const float*)d_in[2];   // event
    const float* w = (const float*)d_in[3];   // weights
    const int n = in_sizes[0];                // 8192

    float* ws       = (float*)d_ws;
    float* rowsum_a = ws;                     // n floats
    float* rowsum_b = ws + n;                 // n floats
    float* sc       = ws + 2 * n;             // 8 floats (4 used)
    float* partials = ws + 2 * n + 8;         // 3 * blocksC floats

    const int blocksA = n / 128;              // 16 rows/wave * 8 waves
    const int blocksC = n / 256;

    disco_pass1   <<<blocksA, 256, 0, stream>>>(x, e, w, rowsum_a, rowsum_b, n);
    disco_stats   <<<1,       256, 0, stream>>>(x, y, w, rowsum_a, rowsum_b, sc, n);
    disco_pass2   <<<blocksC, 256, 0, stream>>>(x, e, w, rowsum_a, rowsum_b, sc, partials, n);
    disco_finalize<<<1,        32, 0, stream>>>(sc, partials, blocksC, (float*)d_out);
}